// GCN_87462714016502
// MI455X (gfx1250) — compile-verified
//
#include <hip/hip_runtime.h>
#include <hip/hip_bf16.h>

typedef __attribute__((ext_vector_type(2))) float v2f;
typedef __attribute__((ext_vector_type(8))) float v8f;

// ---------------------------------------------------------------------------
// Utility fills
// ---------------------------------------------------------------------------
__global__ void gcn_fill_f32(float* __restrict__ p, float val, long long n) {
    long long i = (long long)blockIdx.x * blockDim.x + threadIdx.x;
    if (i < n) p[i] = val;
}

// deg accumulation: one thread per edge, deg pre-initialized to 1.0 (self loop)
__global__ void gcn_deg_accum(const long long* __restrict__ dst, float* __restrict__ deg,
                              long long E) {
    long long e = (long long)blockIdx.x * blockDim.x + threadIdx.x;
    if (e < E) atomicAdd(&deg[(int)dst[e]], 1.0f);
}

// dis = deg^{-1/2} in place (deg >= 1 always due to self-loops)
__global__ void gcn_rsqrt(float* __restrict__ d, int n) {
    int i = blockIdx.x * blockDim.x + threadIdx.x;
    if (i < n) d[i] = rsqrtf(d[i]);
}

// ---------------------------------------------------------------------------
// GEMM1: H[N,32] = X[N,512] @ W[512,32]   (f32 WMMA 16x16x4)
// Block: 128 threads = 4 waves; each wave -> 32 rows x 32 cols (4 accumulators).
// KC=64 chunk of X staged to LDS (coalesced float4); W chunk staged TRANSPOSED
// so each B fragment is one contiguous ds_load_b64.
// ---------------------------------------------------------------------------
#define GCN_TM 128              // rows per block (4 waves x 32 rows)
#define GCN_KC 64               // K chunk
#define GCN_AS (GCN_KC + 4)     // LDS row stride for A tile (pad, even for b64)
#define GCN_BTS (GCN_KC + 2)    // LDS row stride for transposed B tile (66)

__global__ __launch_bounds__(128) void gcn_gemm1(const float* __restrict__ X,
                                                 const float* __restrict__ W,
                                                 float* __restrict__ H,
                                                 int Nrows, int K) {
    __shared__ float sA[GCN_TM * GCN_AS];          // 128 x 68 floats
    __shared__ float sBT[32 * GCN_BTS];            // 32 cols x 66 floats

    const int tid  = threadIdx.x;
    const int wave = tid >> 5;
    const int lane = tid & 31;
    const int rowBase = blockIdx.x * GCN_TM;

    v8f acc00 = {}, acc01 = {};   // rows [0,16) x cols [0,16)/[16,32)
    v8f acc10 = {}, acc11 = {};   // rows [16,32)

    const int aRow0 = wave * 32 + (lane & 15);
    const int aRow1 = aRow0 + 16;
    const int kHalf = (lane >> 4) * 2;     // lanes 0-15 -> K 0,1 ; lanes 16-31 -> K 2,3
    const int col   = lane & 15;

    for (int k0 = 0; k0 < K; k0 += GCN_KC) {
        // stage X tile: 128 rows x 64 k, float4 coalesced (16 vec4 per thread)
        for (int i = tid; i < GCN_TM * (GCN_KC / 4); i += 128) {
            int r  = i / (GCN_KC / 4);
            int c4 = i % (GCN_KC / 4);
            int gr = rowBase + r;
            float4 v = make_float4(0.f, 0.f, 0.f, 0.f);
            if (gr < Nrows) v = *(const float4*)(X + (size_t)gr * K + k0 + c4 * 4);
            int o = r * GCN_AS + c4 * 4;
            sA[o + 0] = v.x; sA[o + 1] = v.y; sA[o + 2] = v.z; sA[o + 3] = v.w;
        }
        // stage W tile transposed: sBT[c][k] = W[k0+k][c]
        for (int i = tid; i < GCN_KC * 8; i += 128) {
            int r  = i / 8;            // k within chunk
            int c4 = i % 8;            // group of 4 cols
            float4 v = *(const float4*)(W + (size_t)(k0 + r) * 32 + c4 * 4);
            int c = c4 * 4;
            sBT[(c + 0) * GCN_BTS + r] = v.x;
            sBT[(c + 1) * GCN_BTS + r] = v.y;
            sBT[(c + 2) * GCN_BTS + r] = v.z;
            sBT[(c + 3) * GCN_BTS + r] = v.w;
        }
        __syncthreads();

        for (int kk = 0; kk < GCN_KC; kk += 4) {
            v2f a0 = *(const v2f*)(sA + aRow0 * GCN_AS + kk + kHalf);
            v2f a1 = *(const v2f*)(sA + aRow1 * GCN_AS + kk + kHalf);
            v2f b0 = *(const v2f*)(sBT + col * GCN_BTS + kk + kHalf);
            v2f b1 = *(const v2f*)(sBT + (col + 16) * GCN_BTS + kk + kHalf);
            acc00 = __builtin_amdgcn_wmma_f32_16x16x4_f32(false, a0, false, b0,
                                                          (short)0, acc00, false, false);
            acc01 = __builtin_amdgcn_wmma_f32_16x16x4_f32(false, a0, false, b1,
                                                          (short)0, acc01, false, false);
            acc10 = __builtin_amdgcn_wmma_f32_16x16x4_f32(false, a1, false, b0,
                                                          (short)0, acc10, false, false);
            acc11 = __builtin_amdgcn_wmma_f32_16x16x4_f32(false, a1, false, b1,
                                                          (short)0, acc11, false, false);
        }
        __syncthreads();
    }

    // C/D layout: VGPR r -> (M = r + 8*(lane>=16), N = lane&15 [+16 for *1])
    const int mOff = (lane >> 4) * 8;
    const int waveRow = rowBase + wave * 32;
    for (int r = 0; r < 8; ++r) {
        int row = waveRow + r + mOff;
        if (row < Nrows) {
            H[(size_t)row * 32 + col]      = acc00[r];
            H[(size_t)row * 32 + col + 16] = acc01[r];
        }
        int row2 = row + 16;
        if (row2 < Nrows) {
            H[(size_t)row2 * 32 + col]      = acc10[r];
            H[(size_t)row2 * 32 + col + 16] = acc11[r];
        }
    }
}

// ---------------------------------------------------------------------------
// GEMM2: H[N,32] = A[N,32] @ W[32,32]   (K=32; W transposed in LDS; A from L2)
// Each wave: 32 rows x 32 cols.
// ---------------------------------------------------------------------------
#define GCN_BTS2 34

__global__ __launch_bounds__(128) void gcn_gemm2(const float* __restrict__ A,
                                                 const float* __restrict__ W,
                                                 float* __restrict__ H,
                                                 int Nrows) {
    __shared__ float sBT[32 * GCN_BTS2];
    const int tid  = threadIdx.x;
    const int wave = tid >> 5;
    const int lane = tid & 31;

    for (int i = tid; i < 32 * 8; i += 128) {
        int r  = i / 8;
        int c4 = i % 8;
        float4 v = *(const float4*)(W + r * 32 + c4 * 4);
        int c = c4 * 4;
        sBT[(c + 0) * GCN_BTS2 + r] = v.x;
        sBT[(c + 1) * GCN_BTS2 + r] = v.y;
        sBT[(c + 2) * GCN_BTS2 + r] = v.z;
        sBT[(c + 3) * GCN_BTS2 + r] = v.w;
    }
    __syncthreads();

    const int row0 = blockIdx.x * GCN_TM + wave * 32;
    if (row0 >= Nrows) return;     // wave-uniform; N % 32 == 0 so tiles are full

    v8f acc00 = {}, acc01 = {}, acc10 = {}, acc11 = {};
    const int aRow0 = row0 + (lane & 15);
    const int aRow1 = aRow0 + 16;
    const int kHalf = (lane >> 4) * 2;
    const int col   = lane & 15;

    for (int kk = 0; kk < 32; kk += 4) {
        v2f a0 = *(const v2f*)(A + (size_t)aRow0 * 32 + kk + kHalf);
        v2f a1 = *(const v2f*)(A + (size_t)aRow1 * 32 + kk + kHalf);
        v2f b0 = *(const v2f*)(sBT + col * GCN_BTS2 + kk + kHalf);
        v2f b1 = *(const v2f*)(sBT + (col + 16) * GCN_BTS2 + kk + kHalf);
        acc00 = __builtin_amdgcn_wmma_f32_16x16x4_f32(false, a0, false, b0,
                                                      (short)0, acc00, false, false);
        acc01 = __builtin_amdgcn_wmma_f32_16x16x4_f32(false, a0, false, b1,
                                                      (short)0, acc01, false, false);
        acc10 = __builtin_amdgcn_wmma_f32_16x16x4_f32(false, a1, false, b0,
                                                      (short)0, acc10, false, false);
        acc11 = __builtin_amdgcn_wmma_f32_16x16x4_f32(false, a1, false, b1,
                                                      (short)0, acc11, false, false);
    }

    const int mOff = (lane >> 4) * 8;
    for (int r = 0; r < 8; ++r) {
        int row = row0 + r + mOff;
        if (row < Nrows) {
            H[(size_t)row * 32 + col]      = acc00[r];
            H[(size_t)row * 32 + col + 16] = acc01[r];
        }
        int row2 = row + 16;
        if (row2 < Nrows) {
            H[(size_t)row2 * 32 + col]      = acc10[r];
            H[(size_t)row2 * 32 + col + 16] = acc11[r];
        }
    }
}

// ---------------------------------------------------------------------------
// Edge scatter: one wave32 per edge, lane = feature.
// out[dst] += h[src] * dis[src]*dis[dst]   (f32 atomics, L2-resident)
// Indices are wave-uniform -> force into SGPRs with readfirstlane.
// ---------------------------------------------------------------------------
__global__ void gcn_scatter(const float* __restrict__ h,
                            const long long* __restrict__ src,
                            const long long* __restrict__ dst,
                            const float* __restrict__ dis,
                            float* __restrict__ out, long long E) {
    long long gtid = (long long)blockIdx.x * blockDim.x + threadIdx.x;
    long long e = gtid >> 5;
    int lane = threadIdx.x & 31;
    if (e >= E) return;
    int s = __builtin_amdgcn_readfirstlane((int)src[e]);
    int d = __builtin_amdgcn_readfirstlane((int)dst[e]);
    float nrm = dis[s] * dis[d];
    float v = h[(size_t)s * 32 + lane] * nrm;
    atomicAdd(&out[(size_t)d * 32 + lane], v);
}

// epilogue 1: out = relu(out + h*dis^2 + b)
__global__ void gcn_epi1(float* __restrict__ out, const float* __restrict__ h,
                         const float* __restrict__ dis, const float* __restrict__ b,
                         int Nrows) {
    long long gtid = (long long)blockIdx.x * blockDim.x + threadIdx.x;
    int row = (int)(gtid >> 5);
    int lane = threadIdx.x & 31;
    if (row >= Nrows) return;
    float di = dis[row];
    size_t o = (size_t)row * 32 + lane;
    float v = out[o] + h[o] * di * di + b[lane];
    out[o] = fmaxf(v, 0.0f);
}

// epilogue 2: v = out + h*dis^2 + b; out = log_softmax(v) across 32 lanes
__global__ void gcn_epi2(float* __restrict__ out, const float* __restrict__ h,
                         const float* __restrict__ dis, const float* __restrict__ b,
                         int Nrows) {
    long long gtid = (long long)blockIdx.x * blockDim.x + threadIdx.x;
    int row = (int)(gtid >> 5);
    int lane = threadIdx.x & 31;
    if (row >= Nrows) return;
    float di = dis[row];
    size_t o = (size_t)row * 32 + lane;
    float v = out[o] + h[o] * di * di + b[lane];
    float m = v;
    for (int off = 16; off > 0; off >>= 1) m = fmaxf(m, __shfl_xor(m, off, 32));
    float ex = expf(v - m);
    float s = ex;
    for (int off = 16; off > 0; off >>= 1) s += __shfl_xor(s, off, 32);
    out[o] = v - m - logf(s);
}

// ---------------------------------------------------------------------------
// Launch
// ---------------------------------------------------------------------------
extern "C" void kernel_launch(void* const* d_in, const int* in_sizes, int n_in,
                              void* d_out, int out_size, void* d_ws, size_t ws_size,
                              hipStream_t stream) {
    const float*      x  = (const float*)d_in[0];
    const long long*  ei = (const long long*)d_in[1];   // int64 [2, E]
    const float*      W1 = (const float*)d_in[2];
    const float*      b1 = (const float*)d_in[3];
    const float*      W2 = (const float*)d_in[4];
    const float*      b2 = (const float*)d_in[5];

    const int Hdim = in_sizes[3];                 // 32
    const int Cdim = in_sizes[5];                 // 32
    const int F    = in_sizes[2] / Hdim;          // 512
    const int N    = in_sizes[0] / F;             // 100000
    const long long E = (long long)in_sizes[1] / 2;

    const long long* srcIdx = ei;
    const long long* dstIdx = ei + E;

    // workspace carve (256B aligned)
    char* ws = (char*)d_ws;
    auto align256 = [](size_t v) { return (v + 255) & ~(size_t)255; };
    float* dis = (float*)ws;                 size_t off = align256((size_t)N * 4);
    float* h1  = (float*)(ws + off);         off += align256((size_t)N * Hdim * 4);
    float* a1  = (float*)(ws + off);         off += align256((size_t)N * Hdim * 4);
    float* h2  = (float*)(ws + off);
    float* a2  = (float*)d_out;              // final aggregation lands in d_out

    const int TPB = 256;
    // 1) deg = 1 (self loop) ; accumulate ; rsqrt
    gcn_fill_f32<<<(N + TPB - 1) / TPB, TPB, 0, stream>>>(dis, 1.0f, (long long)N);
    gcn_deg_accum<<<(unsigned)((E + TPB - 1) / TPB), TPB, 0, stream>>>(dstIdx, dis, E);
    gcn_rsqrt<<<(N + TPB - 1) / TPB, TPB, 0, stream>>>(dis, N);

    // 2) zero aggregation buffers
    gcn_fill_f32<<<(unsigned)(((long long)N * Hdim + TPB - 1) / TPB), TPB, 0, stream>>>(
        a1, 0.0f, (long long)N * Hdim);
    gcn_fill_f32<<<(unsigned)(((long long)N * Cdim + TPB - 1) / TPB), TPB, 0, stream>>>(
        a2, 0.0f, (long long)N * Cdim);

    // 3) layer 1: dense WMMA GEMM, edge scatter, self-loop+bias+relu
    gcn_gemm1<<<(N + GCN_TM - 1) / GCN_TM, 128, 0, stream>>>(x, W1, h1, N, F);
    {
        long long thr = E * 32;
        gcn_scatter<<<(unsigned)((thr + TPB - 1) / TPB), TPB, 0, stream>>>(
            h1, srcIdx, dstIdx, dis, a1, E);
    }
    {
        long long thr = (long long)N * 32;
        gcn_epi1<<<(unsigned)((thr + TPB - 1) / TPB), TPB, 0, stream>>>(a1, h1, dis, b1, N);
    }

    // 4) layer 2: WMMA GEMM (K=32), scatter into d_out, self-loop+bias+log_softmax
    gcn_gemm2<<<(N + GCN_TM - 1) / GCN_TM, 128, 0, stream>>>(a1, W2, h2, N);
    {
        long long thr = E * 32;
        gcn_scatter<<<(unsigned)((thr + TPB - 1) / TPB), TPB, 0, stream>>>(
            h2, srcIdx, dstIdx, dis, a2, E);
    }
    {
        long long thr = (long long)N * 32;
        gcn_epi2<<<(unsigned)((thr + TPB - 1) / TPB), TPB, 0, stream>>>(a2, h2, dis, b2, N);
    }
    (void)n_in; (void)out_size; (void)ws_size;
}